// LuongAttention_83322365542548
// MI455X (gfx1250) — compile-verified
//
#include <hip/hip_runtime.h>
#include <hip/hip_bf16.h>
#include <math.h>

typedef float v2f __attribute__((ext_vector_type(2)));
typedef float v8f __attribute__((ext_vector_type(8)));

#define B_DIM  64
#define S_DIM  4096
#define H_DIM  1024
#define NCHUNK 8
#define SCHUNK (S_DIM / NCHUNK)   // 512 rows per chunk

// workspace layout (in floats)
#define WS_DEC    0
#define WS_SCORES (WS_DEC    + B_DIM * H_DIM)     // 65536
#define WS_PM     (WS_SCORES + B_DIM * S_DIM)     // 327680
#define WS_PL     (WS_PM     + B_DIM * NCHUNK)    // 328192
#define WS_PCTX   (WS_PL     + B_DIM * NCHUNK)    // 328704
// total floats = WS_PCTX + B*NCHUNK*H = 852992  (~3.26 MB)

// ---------------------------------------------------------------------------
// Kernel 1: dec = x @ W^T  via V_WMMA_F32_16X16X4_F32 (exact fp32).
// One wave computes one 16x16 tile of dec; K loop in steps of 4.
// A frag (16x4 f32): lane = M (0..15 twice); VGPR0/1 hold K = {0,1} (lanes 0-15)
//                    and K = {2,3} (lanes 16-31).
// B frag (4x16 f32): lane = N; same K striping pattern.  B[k][n] = W[n][k].
// D frag (16x16 f32): VGPR v -> M = v (lanes 0-15) / v+8 (lanes 16-31), N = lane%16.
// ---------------------------------------------------------------------------
__global__ __launch_bounds__(128)
void proj_wmma_kernel(const float* __restrict__ x, const float* __restrict__ W,
                      float* __restrict__ dec) {
  const int wave = threadIdx.x >> 5;
  const int lane = threadIdx.x & 31;
  const int tile = blockIdx.x * 4 + wave;   // 256 tiles total
  const int m0 = (tile & 3) * 16;           // batch tile (64 rows)
  const int n0 = (tile >> 2) * 16;          // output-feature tile (1024 cols)
  const int mr = lane & 15;
  const int hi = lane >> 4;                 // 0 or 1 (selects K pair)

  const float* arow = x + (m0 + mr) * H_DIM;  // A: x[m][k]
  const float* brow = W + (n0 + mr) * H_DIM;  // B: W[n][k]  (== B[k][n] transposed)

  v8f acc = {};
#if defined(__has_builtin) && __has_builtin(__builtin_amdgcn_wmma_f32_16x16x4_f32)
  for (int k0 = 0; k0 < H_DIM; k0 += 4) {
    v2f a = *reinterpret_cast<const v2f*>(arow + k0 + 2 * hi);
    v2f b = *reinterpret_cast<const v2f*>(brow + k0 + 2 * hi);
    acc = __builtin_amdgcn_wmma_f32_16x16x4_f32(false, a, false, b,
                                                (short)0, acc, false, false);
  }
#else
  // Fallback (should not be taken on gfx1250): scalar FMA producing the same
  // D-fragment layout so the store below stays valid.
  for (int k0 = 0; k0 < H_DIM; ++k0) {
    #pragma unroll
    for (int v = 0; v < 8; ++v) {
      acc[v] += x[(m0 + v + 8 * hi) * H_DIM + k0] * W[(n0 + mr) * H_DIM + k0];
    }
  }
#endif
  #pragma unroll
  for (int v = 0; v < 8; ++v) {
    dec[(m0 + v + 8 * hi) * H_DIM + (n0 + mr)] = acc[v];
  }
}

// ---------------------------------------------------------------------------
// Kernel 2: single streaming pass over encoder_outputs.
// Grid (NCHUNK, B), 256 threads (8 waves).  Each wave owns interleaved s rows,
// lanes split H into 8 coalesced float4 slices.  Online softmax per wave with
// a 32-float context accumulator per lane; wave partials merged in LDS.
// ---------------------------------------------------------------------------
__global__ __launch_bounds__(256)
void attn_stream_kernel(const float* __restrict__ enc, const float* __restrict__ dec,
                        float* __restrict__ scores, float* __restrict__ pm,
                        float* __restrict__ pl, float* __restrict__ pctx) {
  const int b     = blockIdx.y;
  const int chunk = blockIdx.x;
  const int t     = threadIdx.x;
  const int wave  = t >> 5;
  const int lane  = t & 31;

  __shared__ float sdec[H_DIM];
  __shared__ float sctx[H_DIM];
  __shared__ float sm[8];
  __shared__ float sl[8];

  // cache dec[b,:] in LDS (256 threads x float4)
  reinterpret_cast<float4*>(sdec)[t] =
      reinterpret_cast<const float4*>(dec + b * H_DIM)[t];
  __syncthreads();

  const float4* sd4   = reinterpret_cast<const float4*>(sdec);
  const size_t  enc_b = (size_t)b * S_DIM * H_DIM;
  const int     s0    = chunk * SCHUNK;

  float  m_w = -INFINITY;
  float  l_w = 0.f;
  float4 acc[8];
  #pragma unroll
  for (int i = 0; i < 8; ++i) acc[i] = make_float4(0.f, 0.f, 0.f, 0.f);

  for (int j = 0; j < SCHUNK / 8; ++j) {
    const int s = s0 + wave + j * 8;
    const float4* erow =
        reinterpret_cast<const float4*>(enc + enc_b + (size_t)s * H_DIM);

    // prefetch this wave's next row (one cacheline region per lane covers 4KB)
    if (j + 1 < SCHUNK / 8)
      __builtin_prefetch(erow + 8 * (H_DIM / 4) + lane * 8, 0, 1);

    float4 e[8];
    float part = 0.f;
    #pragma unroll
    for (int i = 0; i < 8; ++i) {
      e[i] = erow[i * 32 + lane];
      const float4 d = sd4[i * 32 + lane];
      part += e[i].x * d.x + e[i].y * d.y + e[i].z * d.z + e[i].w * d.w;
    }
    // wave32 all-reduce
    #pragma unroll
    for (int off = 16; off > 0; off >>= 1)
      part += __shfl_xor(part, off, 32);
    const float score = part;
    if (lane == 0) scores[b * S_DIM + s] = score;

    // online softmax update
    const float m_new = fmaxf(m_w, score);
    const float corr  = __expf(m_w - m_new);
    const float p     = __expf(score - m_new);
    l_w = l_w * corr + p;
    #pragma unroll
    for (int i = 0; i < 8; ++i) {
      acc[i].x = acc[i].x * corr + p * e[i].x;
      acc[i].y = acc[i].y * corr + p * e[i].y;
      acc[i].z = acc[i].z * corr + p * e[i].z;
      acc[i].w = acc[i].w * corr + p * e[i].w;
    }
    m_w = m_new;
  }

  // merge 8 wave partials
  if (lane == 0) { sm[wave] = m_w; sl[wave] = l_w; }
  reinterpret_cast<float4*>(sctx)[t] = make_float4(0.f, 0.f, 0.f, 0.f);
  __syncthreads();

  float m_c = sm[0];
  #pragma unroll
  for (int w = 1; w < 8; ++w) m_c = fmaxf(m_c, sm[w]);
  float l_c = 0.f;
  #pragma unroll
  for (int w = 0; w < 8; ++w) l_c += sl[w] * __expf(sm[w] - m_c);

  const float scale = __expf(m_w - m_c);
  #pragma unroll
  for (int i = 0; i < 8; ++i) {
    const int h = i * 128 + lane * 4;
    atomicAdd(&sctx[h + 0], acc[i].x * scale);
    atomicAdd(&sctx[h + 1], acc[i].y * scale);
    atomicAdd(&sctx[h + 2], acc[i].z * scale);
    atomicAdd(&sctx[h + 3], acc[i].w * scale);
  }
  __syncthreads();

  const int pidx = b * NCHUNK + chunk;
  if (t == 0) { pm[pidx] = m_c; pl[pidx] = l_c; }
  reinterpret_cast<float4*>(pctx + (size_t)pidx * H_DIM)[t] =
      reinterpret_cast<const float4*>(sctx)[t];
}

// ---------------------------------------------------------------------------
// Kernel 3: per-batch finalize.  Merge chunk partials (log-sum-exp), write
// context (d_out[0 .. B*H)) and weights (d_out[B*H .. B*H + B*S)).
// ---------------------------------------------------------------------------
__global__ __launch_bounds__(256)
void finalize_kernel(const float* __restrict__ scores, const float* __restrict__ pm,
                     const float* __restrict__ pl, const float* __restrict__ pctx,
                     float* __restrict__ out) {
  const int b = blockIdx.x;
  const int t = threadIdx.x;

  float M = -INFINITY;
  #pragma unroll
  for (int c = 0; c < NCHUNK; ++c) M = fmaxf(M, pm[b * NCHUNK + c]);
  float L = 0.f;
  #pragma unroll
  for (int c = 0; c < NCHUNK; ++c)
    L += pl[b * NCHUNK + c] * __expf(pm[b * NCHUNK + c] - M);
  const float invL = 1.f / L;

  // context: thread t owns h = 4t .. 4t+3
  float4 sum = make_float4(0.f, 0.f, 0.f, 0.f);
  #pragma unroll
  for (int c = 0; c < NCHUNK; ++c) {
    const float sc = __expf(pm[b * NCHUNK + c] - M);
    const float4 v =
        reinterpret_cast<const float4*>(pctx + (size_t)(b * NCHUNK + c) * H_DIM)[t];
    sum.x += v.x * sc; sum.y += v.y * sc; sum.z += v.z * sc; sum.w += v.w * sc;
  }
  sum.x *= invL; sum.y *= invL; sum.z *= invL; sum.w *= invL;
  reinterpret_cast<float4*>(out + b * H_DIM)[t] = sum;

  // weights
  float* outw = out + B_DIM * H_DIM;
  for (int r = 0; r < S_DIM / 256; ++r) {
    const int s = r * 256 + t;
    outw[b * S_DIM + s] = __expf(scores[b * S_DIM + s] - M) * invL;
  }
}

// ---------------------------------------------------------------------------
extern "C" void kernel_launch(void* const* d_in, const int* in_sizes, int n_in,
                              void* d_out, int out_size, void* d_ws, size_t ws_size,
                              hipStream_t stream) {
  (void)in_sizes; (void)n_in; (void)out_size; (void)ws_size;
  const float* dh  = (const float*)d_in[0];   // decoder_hidden [B,H]
  const float* enc = (const float*)d_in[1];   // encoder_outputs [B,S,H]
  const float* W   = (const float*)d_in[2];   // W [H,H]
  float* out = (float*)d_out;                 // context [B,H] ++ weights [B,S]
  float* ws  = (float*)d_ws;

  float* dec    = ws + WS_DEC;
  float* scores = ws + WS_SCORES;
  float* pm     = ws + WS_PM;
  float* pl     = ws + WS_PL;
  float* pctx   = ws + WS_PCTX;

  proj_wmma_kernel<<<dim3(64), dim3(128), 0, stream>>>(dh, W, dec);
  attn_stream_kernel<<<dim3(NCHUNK, B_DIM), dim3(256), 0, stream>>>(
      enc, dec, scores, pm, pl, pctx);
  finalize_kernel<<<dim3(B_DIM), dim3(256), 0, stream>>>(scores, pm, pl, pctx, out);
}